// CrowdGNN_8323646619687
// MI455X (gfx1250) — compile-verified
//
#include <hip/hip_runtime.h>

typedef __attribute__((ext_vector_type(2))) float v2f;
typedef __attribute__((ext_vector_type(8))) float v8f;

#define TPB 256

// DPP16 butterfly add step (row ops stay within 16-lane rows on wave32).
template <int CTRL>
__device__ __forceinline__ float dpp_add(float v) {
  int sh = __builtin_amdgcn_update_dpp(0, __builtin_bit_cast(int, v),
                                       CTRL, 0xF, 0xF, true);
  return v + __builtin_bit_cast(float, sh);
}

// Sum across each 16-lane half-row; result broadcast to all 16 lanes.
__device__ __forceinline__ float row16_sum(float v) {
  v = dpp_add<0xB1>(v);   // quad_perm(1,0,3,2)  : xor 1
  v = dpp_add<0x4E>(v);   // quad_perm(2,3,0,1)  : xor 2
  v = dpp_add<0x141>(v);  // row_half_mirror     : pairs quads within 8
  v = dpp_add<0x140>(v);  // row_mirror          : pairs 8-groups within 16
  return v;
}

// --- deg[i] = 1.0 (self-loop weight) ---------------------------------------
__global__ void k_init_deg(float* __restrict__ deg, int N) {
  int i = blockIdx.x * blockDim.x + threadIdx.x;
  if (i < N) deg[i] = 1.0f;
}

// --- deg[dst] += w  (edge streams non-temporal; deg lives in L2) -----------
__global__ void k_accum_deg(const int* __restrict__ dst, const float* __restrict__ w,
                            float* __restrict__ deg, int E) {
  int e = blockIdx.x * blockDim.x + threadIdx.x;
  if (e < E) {
    int   d  = __builtin_nontemporal_load(dst + e);
    float wv = __builtin_nontemporal_load(w + e);
    atomicAdd(deg + d, wv);
  }
}

// --- dinv = deg > 0 ? rsqrt(deg) : 0 (in place) ----------------------------
__global__ void k_deg_to_dinv(float* __restrict__ deg, int N) {
  int i = blockIdx.x * blockDim.x + threadIdx.x;
  if (i < N) {
    float dv = deg[i];
    deg[i] = (dv > 0.0f) ? rsqrtf(dv) : 0.0f;
  }
}

// --- agg[i][:] = dinv[i]^2 * x[i][:]  (self-loop contribution) -------------
__global__ void k_init_agg(const float* __restrict__ x, const float* __restrict__ dinv,
                           float* __restrict__ agg, int N) {
  int i = blockIdx.x * blockDim.x + threadIdx.x;
  if (i < N) {
    float di = dinv[i];
    float nrm = di * di;
    float4 xv = reinterpret_cast<const float4*>(x)[i];
    float4 av;
    av.x = xv.x * nrm; av.y = xv.y * nrm; av.z = xv.z * nrm; av.w = xv.w * nrm;
    reinterpret_cast<float4*>(agg)[i] = av;
  }
}

// --- conv1 aggregation: agg[dst][:] += x[src][:] * norm --------------------
__global__ void k_scatter1(const int* __restrict__ src, const int* __restrict__ dst,
                           const float* __restrict__ w, const float* __restrict__ dinv,
                           const float* __restrict__ x, float* __restrict__ agg, int E) {
  int e = blockIdx.x * blockDim.x + threadIdx.x;
  if (e < E) {
    int   s  = __builtin_nontemporal_load(src + e);
    int   d  = __builtin_nontemporal_load(dst + e);
    float wv = __builtin_nontemporal_load(w + e);
    float norm = dinv[s] * wv * dinv[d];
    float4 xs = reinterpret_cast<const float4*>(x)[s];
    float* ap = agg + (size_t)d * 4;
    atomicAdd(ap + 0, xs.x * norm);
    atomicAdd(ap + 1, xs.y * norm);
    atomicAdd(ap + 2, xs.z * norm);
    atomicAdd(ap + 3, xs.w * norm);
  }
}

// --- fused per-node transform via WMMA:
//     h(16x16) = agg_tile(16x4) @ W1(4x16)   [V_WMMA_F32_16X16X4_F32]
//     t[node]  = relu(h + b1) @ W2           (DPP row reduction, no LDS)
__global__ void k_transform_wmma(const float* __restrict__ agg,
                                 const float* __restrict__ W1,
                                 const float* __restrict__ b1,
                                 const float* __restrict__ W2,
                                 float* __restrict__ t, int N, int nTiles) {
  int lane  = threadIdx.x & 31;
  int half  = lane >> 4;      // A: 0 -> K=0..1, 1 -> K=2..3 ;  D rows: M+0 / M+8
  int l16   = lane & 15;
  int wave  = (blockIdx.x * blockDim.x + threadIdx.x) >> 5;
  int nWaves = (gridDim.x * blockDim.x) >> 5;

  // B operand (4x16, K-major): lane<16 holds K=0,1 of col l16; lane>=16 holds K=2,3
  v2f B;
  B.x = W1[(half * 2 + 0) * 16 + l16];
  B.y = W1[(half * 2 + 1) * 16 + l16];
  float bias = b1[l16];   // feature n = l16 per lane
  float w2v  = W2[l16];

  for (int tile = wave; tile < nTiles; tile += nWaves) {
    int base = tile * 16;
    int m = base + l16;
    int mc = (m < N) ? m : (N - 1);          // clamp tail loads (stores guarded)
    // A operand (16x4): lane<16 holds agg[m][0:2], lane>=16 holds agg[m][2:4]
    const float* arow = agg + (size_t)mc * 4 + half * 2;
    v2f A;
    A.x = arow[0];
    A.y = arow[1];
    v8f C = {};
    C = __builtin_amdgcn_wmma_f32_16x16x4_f32(
        /*neg_a=*/false, A, /*neg_b=*/false, B,
        /*c_mod=*/(short)0, C, /*reuse_a=*/false, /*reuse_b=*/false);

    // Per accumulator r: h[base + half*8 + r][n=l16] lives in C[r].
    // t = sum_n relu(h+b1[n]) * W2[n] -> 16-lane DPP reduction, then select
    // the r matching this lane so a single coalesced store writes 16 nodes.
    float res = 0.0f;
#pragma unroll
    for (int r = 0; r < 8; ++r) {
      float h = fmaxf(C[r] + bias, 0.0f);
      float v = row16_sum(h * w2v);          // broadcast across the half-row
      res = (l16 == r) ? v : res;            // v_cndmask, no branch
    }
    int node = base + half * 8 + l16;        // lanes 0-7 and 16-23 are writers
    if (l16 < 8 && node < N) t[node] = res;
  }
}

// --- out[i] = dinv[i]^2 * t[i] + b2 (self-loop + bias) ---------------------
__global__ void k_init_out(const float* __restrict__ t, const float* __restrict__ dinv,
                           const float* __restrict__ b2, float* __restrict__ out, int N) {
  int i = blockIdx.x * blockDim.x + threadIdx.x;
  if (i < N) {
    float di = dinv[i];
    out[i] = di * di * t[i] + b2[0];
  }
}

// --- conv2 aggregation: out[dst] += t[src] * norm --------------------------
__global__ void k_scatter2(const int* __restrict__ src, const int* __restrict__ dst,
                           const float* __restrict__ w, const float* __restrict__ dinv,
                           const float* __restrict__ t, float* __restrict__ out, int E) {
  int e = blockIdx.x * blockDim.x + threadIdx.x;
  if (e < E) {
    int   s  = __builtin_nontemporal_load(src + e);
    int   d  = __builtin_nontemporal_load(dst + e);
    float wv = __builtin_nontemporal_load(w + e);
    float norm = dinv[s] * wv * dinv[d];
    atomicAdd(out + d, t[s] * norm);
  }
}

extern "C" void kernel_launch(void* const* d_in, const int* in_sizes, int n_in,
                              void* d_out, int out_size, void* d_ws, size_t ws_size,
                              hipStream_t stream) {
  const float* x  = (const float*)d_in[0];   // [N,4]
  const int*   ei = (const int*)d_in[1];     // [2,E] row-major (int32)
  const float* w  = (const float*)d_in[2];   // [E]
  const float* W1 = (const float*)d_in[3];   // [4,16]
  const float* b1 = (const float*)d_in[4];   // [16]
  const float* W2 = (const float*)d_in[5];   // [16,1]
  const float* b2 = (const float*)d_in[6];   // [1]

  const int N = in_sizes[0] / 4;
  const int E = in_sizes[2];
  const int* src = ei;
  const int* dst = ei + E;

  // workspace: dinv [N] | agg [4N] | t [N]  (6N floats = 12 MB)
  float* dinv = (float*)d_ws;
  float* agg  = dinv + N;
  float* t    = agg + (size_t)4 * N;
  float* out  = (float*)d_out;

  const int nb = (N + TPB - 1) / TPB;
  const int eb = (E + TPB - 1) / TPB;
  const int nTiles = (N + 15) / 16;
  const int wavesPerBlock = TPB / 32;
  const int tb = (nTiles + wavesPerBlock - 1) / wavesPerBlock;

  k_init_deg      <<<nb, TPB, 0, stream>>>(dinv, N);
  k_accum_deg     <<<eb, TPB, 0, stream>>>(dst, w, dinv, E);
  k_deg_to_dinv   <<<nb, TPB, 0, stream>>>(dinv, N);
  k_init_agg      <<<nb, TPB, 0, stream>>>(x, dinv, agg, N);
  k_scatter1      <<<eb, TPB, 0, stream>>>(src, dst, w, dinv, x, agg, E);
  k_transform_wmma<<<tb, TPB, 0, stream>>>(agg, W1, b1, W2, t, N, nTiles);
  k_init_out      <<<nb, TPB, 0, stream>>>(t, dinv, b2, out, N);
  k_scatter2      <<<eb, TPB, 0, stream>>>(src, dst, w, dinv, t, out, E);
}